// InteractionBlock_9131100471465
// MI455X (gfx1250) — compile-verified
//
#include <hip/hip_runtime.h>
#include <cstdint>

// CDNA5 / gfx1250, wave32. All matrix math via V_WMMA_F32_16X16X4_F32.

typedef float v2f __attribute__((ext_vector_type(2)));
typedef float v8f __attribute__((ext_vector_type(8)));

#define TSTR 68    // LDS row stride for 16x64 tiles (padded: bank = 4*row + col, distinct)
#define WSTR 68    // LDS row stride for 64x64 weight tiles
#define XSTR 260   // LDS row stride for 16x256 tiles

__device__ __forceinline__ v8f wmma4(v2f a, v2f b, v8f c) {
  // D = A(16x4,f32) * B(4x16,f32) + C(16x16,f32)
  return __builtin_amdgcn_wmma_f32_16x16x4_f32(false, a, false, b, (short)0, c,
                                               false, false);
}

// single v_rcp_f32 instead of the IEEE div expansion (1-ulp, << GEMM noise)
__device__ __forceinline__ float silu_f(float y) {
  return y * __builtin_amdgcn_rcpf(1.0f + __expf(-y));
}

// gfx1250 hardware transcendental V_TANH_F32
__device__ __forceinline__ float tanh_hw(float x) {
  float r;
  asm("v_tanh_f32 %0, %1" : "=v"(r) : "v"(x));
  return r;
}

// ---------------------------------------------------------------- zero agg
__global__ void zero_f4(float4* __restrict__ p, long n4) {
  long i = (long)blockIdx.x * blockDim.x + threadIdx.x;
  long st = (long)gridDim.x * blockDim.x;
  float4 z = make_float4(0.f, 0.f, 0.f, 0.f);
  for (; i < n4; i += st) p[i] = z;
}

// ------------------------------------------------- K1: x_down = silu(x@Wd+b)
__global__ void __launch_bounds__(128) down_kernel(
    const float* __restrict__ x, const float* __restrict__ Wd,
    const float* __restrict__ bd, float* __restrict__ xd, int N) {
  extern __shared__ float sm[];
  const int tid = threadIdx.x;
  const int base = blockIdx.x * 16;

  for (int i = tid; i < 16 * 256; i += 128) {
    int r = i >> 8, cc = i & 255;
    int row = base + r; if (row >= N) row = N - 1;
    sm[r * XSTR + cc] = x[(size_t)row * 256 + cc];
  }
  __syncthreads();

  const int wave = tid >> 5, lane = tid & 31, lm = lane & 15, hl = lane >> 4;
  const int col0 = wave * 16;
  v8f c = {};
  for (int k0 = 0; k0 < 256; k0 += 4) {
    const int kr = k0 + 2 * hl;
    v2f a, b;
    a.x = sm[lm * XSTR + kr];      a.y = sm[lm * XSTR + kr + 1];
    b.x = Wd[kr * 64 + col0 + lm]; b.y = Wd[(kr + 1) * 64 + col0 + lm];
    c = wmma4(a, b, c);
  }
  const int col = col0 + lm;
  const float bb = bd[col];
  if (base + 16 <= N) {  // full tile: base ptr + immediate offsets, no branches
    float* p = xd + (size_t)(base + 8 * hl) * 64 + col;
#pragma unroll
    for (int v = 0; v < 8; ++v) p[v * 64] = silu_f(c[v] + bb);
  } else {
#pragma unroll
    for (int v = 0; v < 8; ++v) {
      int row = base + v + 8 * hl;
      if (row < N) xd[(size_t)row * 64 + col] = silu_f(c[v] + bb);
    }
  }
}

// ------------------------------------------------- K2: fused edge pipeline
__global__ void __launch_bounds__(256) edge_kernel(
    const float* __restrict__ xd, const int* __restrict__ ei,
    const float* __restrict__ ea, const float* __restrict__ ax,
    const float* __restrict__ Wm, const float* __restrict__ bm,
    const float* __restrict__ gm, const float* __restrict__ btm,
    const float* __restrict__ mm, const float* __restrict__ vm,
    const float* __restrict__ Ah,
    const float* __restrict__ We, const float* __restrict__ be,
    const float* __restrict__ ge, const float* __restrict__ bte,
    const float* __restrict__ me, const float* __restrict__ ve,
    float* __restrict__ agg, float* __restrict__ eout, int E, int numTiles) {
  extern __shared__ float sm[];
  const int tid = threadIdx.x;

  // stage weights once per block: Wm, A[0..3], We (padded rows)
  {
    const float* srcs[6] = {Wm, Ah, Ah + 4096, Ah + 8192, Ah + 12288, We};
#pragma unroll 1
    for (int i = tid; i < 6 * 4096; i += 256) {
      int mtx = i >> 12, idx = i & 4095, k = idx >> 6, n = idx & 63;
      sm[mtx * (64 * WSTR) + k * WSTR + n] = srcs[mtx][idx];
    }
  }
  __syncthreads();

  const float* wWm = sm;
  const float* wWe = sm + 5 * (64 * WSTR);
  const int wave = tid >> 5, lane = tid & 31, lm = lane & 15, hl = lane >> 4;
  float* tb  = sm + 6 * (64 * WSTR) + wave * (4 * 16 * TSTR + 16);
  float* xi  = tb;
  float* xj  = tb + 16 * TSTR;
  float* tm  = tb + 2 * 16 * TSTR;  // t, then overwritten with m
  float* wt  = tb + 3 * 16 * TSTR;  // edge_attr * edge_aux
  int*  dstT = (int*)(tb + 4 * 16 * TSTR);

  // ---- tile-invariant BN constants (per-lane column = nt*16+lm)
  float sMc[4], shMc[4], bbMc[4], sEc[4], shEc[4], bbEc[4];
#pragma unroll
  for (int nt = 0; nt < 4; ++nt) {
    const int col = nt * 16 + lm;
    float s = gm[col] * rsqrtf(vm[col] + 1e-5f);
    sMc[nt] = s; shMc[nt] = btm[col] - mm[col] * s; bbMc[nt] = bm[col];
    s = ge[col] * rsqrtf(ve[col] + 1e-5f);
    sEc[nt] = s; shEc[nt] = bte[col] - me[col] * s; bbEc[nt] = be[col];
  }

  const int gw = blockIdx.x * 8 + wave;
  const int nw = gridDim.x * 8;
#pragma unroll 1
  for (int tile = gw; tile < numTiles; tile += nw) {
    const int base = tile * 16;
    const bool full = (base + 16 <= E);

    // prefetch next tile's streaming operands (the HBM-bound streams)
    {
      const int nb = base + nw * 16;
      if (nb < E) {
        __builtin_prefetch(ea + (size_t)nb * 64 + 2 * lane, 0, 1);
        __builtin_prefetch(ax + (size_t)nb * 64 + 2 * lane, 0, 1);
      }
    }

    // ---- stage 16-edge tile: gathers hit L2 (x_down is 12.8MB, resident)
#pragma unroll 1
    for (int r = 0; r < 16; ++r) {
      int e = base + r; if (e >= E) e = E - 1;
      int s = ei[e], d = ei[E + e];
      if (lane == 0) dstT[r] = d;
      float2 vi = *(const float2*)(xd + (size_t)s * 64 + 2 * lane);
      float2 vj = *(const float2*)(xd + (size_t)d * 64 + 2 * lane);
      float2 va = *(const float2*)(ea + (size_t)e * 64 + 2 * lane);
      float2 vx = *(const float2*)(ax + (size_t)e * 64 + 2 * lane);
      *(float2*)(xi + r * TSTR + 2 * lane) = vi;
      *(float2*)(xj + r * TSTR + 2 * lane) = vj;
      float2 t2; t2.x = vi.x * vj.x * va.x; t2.y = vi.y * vj.y * va.y;
      *(float2*)(tm + r * TSTR + 2 * lane) = t2;
      float2 w2; w2.x = va.x * vx.x; w2.y = va.y * vx.y;
      *(float2*)(wt + r * TSTR + 2 * lane) = w2;
    }

    // ---- m = silu(bn(t @ Wm + bm))
    v8f mfrag[4];
#pragma unroll
    for (int nt = 0; nt < 4; ++nt) {
      v8f c = {};
      for (int k0 = 0; k0 < 64; k0 += 4) {
        const int kr = k0 + 2 * hl;
        v2f a, b;
        a.x = tm[lm * TSTR + kr];            a.y = tm[lm * TSTR + kr + 1];
        b.x = wWm[kr * WSTR + nt * 16 + lm]; b.y = wWm[(kr + 1) * WSTR + nt * 16 + lm];
        c = wmma4(a, b, c);
      }
#pragma unroll
      for (int v = 0; v < 8; ++v)
        c[v] = silu_f((c[v] + bbMc[nt]) * sMc[nt] + shMc[nt]);
      mfrag[nt] = c;
    }
    // overwrite t with m (A-matrix source for the We GEMM)
#pragma unroll
    for (int nt = 0; nt < 4; ++nt)
#pragma unroll
      for (int v = 0; v < 8; ++v)
        tm[(v + 8 * hl) * TSTR + nt * 16 + lm] = mfrag[nt][v];

    // per-tile, head-invariant values: dst rows + score weights
    int dstv[8];
    float q[8][4];
#pragma unroll
    for (int v = 0; v < 8; ++v) {
      dstv[v] = dstT[v + 8 * hl];
#pragma unroll
      for (int nt = 0; nt < 4; ++nt)
        q[v][nt] = wt[(v + 8 * hl) * TSTR + nt * 16 + lm];
    }

    // ---- heads: ai = x_i@A_h, aj = x_j@A_h, tanhshrink score, scatter
#pragma unroll 1
    for (int h = 0; h < 4; ++h) {
      const float* WA = sm + (1 + h) * (64 * WSTR);
      v8f cai[4], caj[4];
#pragma unroll
      for (int nt = 0; nt < 4; ++nt) {
        v8f ci = {}, cj = {};
        for (int k0 = 0; k0 < 64; k0 += 4) {
          const int kr = k0 + 2 * hl;
          v2f aiF, ajF, b;
          aiF.x = xi[lm * TSTR + kr]; aiF.y = xi[lm * TSTR + kr + 1];
          ajF.x = xj[lm * TSTR + kr]; ajF.y = xj[lm * TSTR + kr + 1];
          b.x = WA[kr * WSTR + nt * 16 + lm];
          b.y = WA[(kr + 1) * WSTR + nt * 16 + lm];
          ci = wmma4(aiF, b, ci);
          cj = wmma4(ajF, b, cj);
        }
        cai[nt] = ci; caj[nt] = cj;
      }
      // tanhshrink(mean over 64 cols of ai*aj*w)
      float alphav[8];
#pragma unroll
      for (int v = 0; v < 8; ++v) {
        float p = 0.f;
#pragma unroll
        for (int nt = 0; nt < 4; ++nt)
          p += cai[nt][v] * caj[nt][v] * q[v][nt];
        p += __shfl_xor(p, 1, 16);
        p += __shfl_xor(p, 2, 16);
        p += __shfl_xor(p, 4, 16);
        p += __shfl_xor(p, 8, 16);
        const float sc = p * (1.0f / 64.0f);
        alphav[v] = sc - tanh_hw(sc);
      }
      if (full) {
#pragma unroll
        for (int v = 0; v < 8; ++v) {
          float* dp = agg + (size_t)dstv[v] * 256 + h * 64 + lm;
#pragma unroll
          for (int nt = 0; nt < 4; ++nt)
            atomicAdd(dp + nt * 16, alphav[v] * mfrag[nt][v]);  // L2-resident
        }
      } else {
#pragma unroll
        for (int v = 0; v < 8; ++v) {
          if (base + v + 8 * hl < E) {
            float* dp = agg + (size_t)dstv[v] * 256 + h * 64 + lm;
#pragma unroll
            for (int nt = 0; nt < 4; ++nt)
              atomicAdd(dp + nt * 16, alphav[v] * mfrag[nt][v]);
          }
        }
      }
    }

    // ---- e = silu(bn(m @ We + be))
#pragma unroll
    for (int nt = 0; nt < 4; ++nt) {
      v8f c = {};
      for (int k0 = 0; k0 < 64; k0 += 4) {
        const int kr = k0 + 2 * hl;
        v2f a, b;
        a.x = tm[lm * TSTR + kr];            a.y = tm[lm * TSTR + kr + 1];
        b.x = wWe[kr * WSTR + nt * 16 + lm]; b.y = wWe[(kr + 1) * WSTR + nt * 16 + lm];
        c = wmma4(a, b, c);
      }
      const int col = nt * 16 + lm;
      if (full) {
        float* p = eout + (size_t)(base + 8 * hl) * 64 + col;
#pragma unroll
        for (int v = 0; v < 8; ++v)
          p[v * 64] = silu_f((c[v] + bbEc[nt]) * sEc[nt] + shEc[nt]);
      } else {
#pragma unroll
        for (int v = 0; v < 8; ++v) {
          const int gr = base + v + 8 * hl;
          if (gr < E)
            eout[(size_t)gr * 64 + col] =
                silu_f((c[v] + bbEc[nt]) * sEc[nt] + shEc[nt]);
        }
      }
    }
  }
}

// ---------------------------------- K3: out = x + silu(bn(agg@Wh + bh))
__global__ void __launch_bounds__(128) out_kernel(
    const float* __restrict__ aggm, const float* __restrict__ x,
    const float* __restrict__ Wh, const float* __restrict__ bh,
    const float* __restrict__ gh, const float* __restrict__ bth,
    const float* __restrict__ mh, const float* __restrict__ vh,
    float* __restrict__ outx, int N) {
  extern __shared__ float sm[];
  const int tid = threadIdx.x;
  const int base = blockIdx.x * 16;
  for (int i = tid; i < 16 * 256; i += 128) {
    int r = i >> 8, cc = i & 255;
    int row = base + r; if (row >= N) row = N - 1;
    sm[r * XSTR + cc] = aggm[(size_t)row * 256 + cc];
  }
  __syncthreads();
  const int wave = tid >> 5, lane = tid & 31, lm = lane & 15, hl = lane >> 4;
#pragma unroll 1
  for (int nt = 0; nt < 4; ++nt) {
    const int col = wave * 64 + nt * 16 + lm;
    v8f c = {};
    for (int k0 = 0; k0 < 256; k0 += 4) {
      const int kr = k0 + 2 * hl;
      v2f a, b;
      a.x = sm[lm * XSTR + kr];         a.y = sm[lm * XSTR + kr + 1];
      b.x = Wh[(size_t)kr * 256 + col]; b.y = Wh[(size_t)(kr + 1) * 256 + col];
      c = wmma4(a, b, c);
    }
    const float s  = gh[col] * rsqrtf(vh[col] + 1e-5f);
    const float sh = bth[col] - mh[col] * s;
    const float bb = bh[col];
    if (base + 16 <= N) {
      const float* xp = x + (size_t)(base + 8 * hl) * 256 + col;
      float* op = outx + (size_t)(base + 8 * hl) * 256 + col;
#pragma unroll
      for (int v = 0; v < 8; ++v)
        op[v * 256] = xp[v * 256] + silu_f((c[v] + bb) * s + sh);
    } else {
#pragma unroll
      for (int v = 0; v < 8; ++v) {
        const int row = base + v + 8 * hl;
        if (row < N)
          outx[(size_t)row * 256 + col] =
              x[(size_t)row * 256 + col] + silu_f((c[v] + bb) * s + sh);
      }
    }
  }
}

// --------------------------------------------------------------- launcher
extern "C" void kernel_launch(void* const* d_in, const int* in_sizes, int n_in,
                              void* d_out, int out_size, void* d_ws,
                              size_t ws_size, hipStream_t stream) {
  const float* x   = (const float*)d_in[0];
  const float* ea  = (const float*)d_in[1];
  const int*   ei  = (const int*)d_in[2];   // JAX x64-off: int32
  const float* ax  = (const float*)d_in[3];
  const float* Wd  = (const float*)d_in[4];
  const float* bd  = (const float*)d_in[5];
  const float* Wm  = (const float*)d_in[6];
  const float* bm  = (const float*)d_in[7];
  const float* gm  = (const float*)d_in[8];
  const float* btm = (const float*)d_in[9];
  const float* mm  = (const float*)d_in[10];
  const float* vm  = (const float*)d_in[11];
  const float* Ah  = (const float*)d_in[12];
  const float* Wh  = (const float*)d_in[13];
  const float* bh  = (const float*)d_in[14];
  const float* gh  = (const float*)d_in[15];
  const float* bth = (const float*)d_in[16];
  const float* mh  = (const float*)d_in[17];
  const float* vh  = (const float*)d_in[18];
  const float* We  = (const float*)d_in[19];
  const float* be  = (const float*)d_in[20];
  const float* ge  = (const float*)d_in[21];
  const float* bte = (const float*)d_in[22];
  const float* me  = (const float*)d_in[23];
  const float* ve  = (const float*)d_in[24];

  const int N = in_sizes[0] / 256;
  const int E = in_sizes[1] / 64;

  float* xd   = (float*)d_ws;            // [N,64]
  float* agg  = xd + (size_t)N * 64;     // [N,256] (= concat over heads)
  float* outx = (float*)d_out;           // [N,256]
  float* eout = outx + (size_t)N * 256;  // [E,64]

  {  // zero the L2-resident accumulator every launch (ws is poisoned)
    long n4 = ((long)N * 256) / 4;
    int grd = (int)((n4 + 4095) / 4096);
    if (grd < 1) grd = 1; if (grd > 8192) grd = 8192;
    zero_f4<<<grd, 256, 0, stream>>>((float4*)agg, n4);
  }
  {  // K1
    int tiles = (N + 15) / 16;
    down_kernel<<<tiles, 128, (size_t)16 * XSTR * 4, stream>>>(x, Wd, bd, xd, N);
  }
  {  // K2
    int numTiles = (E + 15) / 16;
    int blocks = (numTiles + 7) / 8;
    if (blocks > 1536) blocks = 1536;
    size_t shbytes = (size_t)(6 * 64 * WSTR + 8 * (4 * 16 * TSTR + 16)) * 4;
    hipFuncSetAttribute(reinterpret_cast<const void*>(edge_kernel),
                        hipFuncAttributeMaxDynamicSharedMemorySize, (int)shbytes);
    edge_kernel<<<blocks, 256, shbytes, stream>>>(
        xd, ei, ea, ax, Wm, bm, gm, btm, mm, vm, Ah, We, be, ge, bte, me, ve,
        agg, eout, E, numTiles);
  }
  {  // K3
    int tiles = (N + 15) / 16;
    out_kernel<<<tiles, 128, (size_t)16 * XSTR * 4, stream>>>(
        agg, x, Wh, bh, gh, bth, mh, vh, outx, N);
  }
}